// MyNet_17901423689818
// MI455X (gfx1250) — compile-verified
//
#include <hip/hip_runtime.h>
#include <hip/hip_bf16.h>
#include <math.h>

#define N_NODES 1024
#define DMODEL  128
#define NHEAD   8
#define HDIM    16
#define DFF     512
#define TOPK    5

typedef __attribute__((ext_vector_type(2))) float v2f;
typedef __attribute__((ext_vector_type(8))) float v8f;

#if defined(__has_builtin)
#if __has_builtin(__builtin_amdgcn_sched_group_barrier)
#define SCHED_LOADS_THEN_WMMA(nload, nwmma)                      \
  do {                                                           \
    __builtin_amdgcn_sched_group_barrier(0x020, (nload), 0);     \
    __builtin_amdgcn_sched_group_barrier(0x008, (nwmma), 0);     \
  } while (0)
#endif
#endif
#ifndef SCHED_LOADS_THEN_WMMA
#define SCHED_LOADS_THEN_WMMA(nload, nwmma) do {} while (0)
#endif

// ---------------------------------------------------------------------------
// fp32 WMMA GEMM:  C[M,Nc] = act(A[M,K] @ W[Nc,K]^T + bias[Nc])
// One wave computes a 16x64 tile of C (4 accumulators); the K-loop is
// software-pipelined: fragments for step k+1 are loaded before the 4
// V_WMMA_F32_16X16X4_F32 issues that consume step k, so WMMAs only wait on
// loads issued a full iteration earlier.
// grid = (Nc/64, M/16), block = 32.
// ---------------------------------------------------------------------------
__global__ void k_gemm_nt(const float* __restrict__ A, const float* __restrict__ W,
                          const float* __restrict__ bias, float* __restrict__ C,
                          int M, int Nc, int K, int relu) {
  const int nt   = blockIdx.x * 64;
  const int mt   = blockIdx.y * 16;
  const int lane = threadIdx.x & 31;
  const int half = lane >> 4;   // 0: K sub {0,1} ; 1: K sub {2,3}
  const int l    = lane & 15;
  const float* __restrict__ arow = A + (size_t)(mt + l) * K + 2 * half;
  const float* __restrict__ w0   = W + (size_t)(nt + l) * K + 2 * half;
  const float* __restrict__ w1   = w0 + (size_t)16 * K;
  const float* __restrict__ w2   = w0 + (size_t)32 * K;
  const float* __restrict__ w3   = w0 + (size_t)48 * K;
  v8f acc0 = {}, acc1 = {}, acc2 = {}, acc3 = {};

  v2f a_c  = *(const v2f*)(arow);
  v2f b0_c = *(const v2f*)(w0);
  v2f b1_c = *(const v2f*)(w1);
  v2f b2_c = *(const v2f*)(w2);
  v2f b3_c = *(const v2f*)(w3);
  for (int k0 = 0; k0 + 4 < K; k0 += 4) {
    v2f a_n  = *(const v2f*)(arow + k0 + 4);
    v2f b0_n = *(const v2f*)(w0 + k0 + 4);
    v2f b1_n = *(const v2f*)(w1 + k0 + 4);
    v2f b2_n = *(const v2f*)(w2 + k0 + 4);
    v2f b3_n = *(const v2f*)(w3 + k0 + 4);
    acc0 = __builtin_amdgcn_wmma_f32_16x16x4_f32(false, a_c, false, b0_c, (short)0, acc0, false, false);
    acc1 = __builtin_amdgcn_wmma_f32_16x16x4_f32(false, a_c, false, b1_c, (short)0, acc1, false, false);
    acc2 = __builtin_amdgcn_wmma_f32_16x16x4_f32(false, a_c, false, b2_c, (short)0, acc2, false, false);
    acc3 = __builtin_amdgcn_wmma_f32_16x16x4_f32(false, a_c, false, b3_c, (short)0, acc3, false, false);
    SCHED_LOADS_THEN_WMMA(5, 4);
    a_c = a_n; b0_c = b0_n; b1_c = b1_n; b2_c = b2_n; b3_c = b3_n;
  }
  acc0 = __builtin_amdgcn_wmma_f32_16x16x4_f32(false, a_c, false, b0_c, (short)0, acc0, false, false);
  acc1 = __builtin_amdgcn_wmma_f32_16x16x4_f32(false, a_c, false, b1_c, (short)0, acc1, false, false);
  acc2 = __builtin_amdgcn_wmma_f32_16x16x4_f32(false, a_c, false, b2_c, (short)0, acc2, false, false);
  acc3 = __builtin_amdgcn_wmma_f32_16x16x4_f32(false, a_c, false, b3_c, (short)0, acc3, false, false);

  v8f accs[4] = {acc0, acc1, acc2, acc3};
#pragma unroll
  for (int s = 0; s < 4; ++s) {
    const int n = nt + 16 * s + l;
    const float bv = bias[n];
#pragma unroll
    for (int r = 0; r < 8; ++r) {
      int m = mt + half * 8 + r;
      float val = accs[s][r] + bv;
      if (relu) val = fmaxf(val, 0.0f);
      C[(size_t)m * Nc + n] = val;
    }
  }
}

// ---------------------------------------------------------------------------
// Per-head edge-bias scalars: alpha[h] = we[h]·edge_w ; beta[h] = we[h]·edge_b + be[h]
// ---------------------------------------------------------------------------
__global__ void k_headcoef(const float* __restrict__ we, const float* __restrict__ be,
                           const float* __restrict__ ew, const float* __restrict__ eb,
                           float* __restrict__ hc) {
  int h = threadIdx.x;
  if (h < NHEAD) {
    float a = 0.f, b = 0.f;
    for (int d = 0; d < DMODEL; ++d) {
      float w = we[h * DMODEL + d];
      a += w * ew[d];
      b += w * eb[d];
    }
    hc[h] = a;
    hc[NHEAD + h] = b + be[h];
  }
}

// ---------------------------------------------------------------------------
// Attention scores for one head via fp32 WMMA, 16(i) x 64(j) tile per wave:
//   S[i,j] = (q_i . k_j)/4 + (adjm[i,j] > 0 ? alpha[h] : 0) + beta[h]
// K-depth is only 16, so all 20 fragments are loaded up front, then the 16
// WMMAs are issued back-to-back.
// grid = (N/64, N/16), block = 32.
// ---------------------------------------------------------------------------
__global__ void k_scores(const float* __restrict__ q, const float* __restrict__ kk,
                         const float* __restrict__ adjm, const float* __restrict__ hc,
                         float* __restrict__ S, int head) {
  const int jt   = blockIdx.x * 64;
  const int it   = blockIdx.y * 16;
  const int lane = threadIdx.x & 31;
  const int half = lane >> 4;
  const int l    = lane & 15;
  const float* __restrict__ qrow = q  + (size_t)(it + l) * DMODEL + head * HDIM + 2 * half;
  const float* __restrict__ k0r  = kk + (size_t)(jt + l) * DMODEL + head * HDIM + 2 * half;
  const float* __restrict__ k1r  = k0r + (size_t)16 * DMODEL;
  const float* __restrict__ k2r  = k0r + (size_t)32 * DMODEL;
  const float* __restrict__ k3r  = k0r + (size_t)48 * DMODEL;

  v2f a[4], b0[4], b1[4], b2[4], b3[4];
#pragma unroll
  for (int ks = 0; ks < 4; ++ks) {
    a[ks]  = *(const v2f*)(qrow + 4 * ks);
    b0[ks] = *(const v2f*)(k0r + 4 * ks);
    b1[ks] = *(const v2f*)(k1r + 4 * ks);
    b2[ks] = *(const v2f*)(k2r + 4 * ks);
    b3[ks] = *(const v2f*)(k3r + 4 * ks);
  }
  v8f acc0 = {}, acc1 = {}, acc2 = {}, acc3 = {};
#pragma unroll
  for (int ks = 0; ks < 4; ++ks) {
    acc0 = __builtin_amdgcn_wmma_f32_16x16x4_f32(false, a[ks], false, b0[ks], (short)0, acc0, false, false);
    acc1 = __builtin_amdgcn_wmma_f32_16x16x4_f32(false, a[ks], false, b1[ks], (short)0, acc1, false, false);
    acc2 = __builtin_amdgcn_wmma_f32_16x16x4_f32(false, a[ks], false, b2[ks], (short)0, acc2, false, false);
    acc3 = __builtin_amdgcn_wmma_f32_16x16x4_f32(false, a[ks], false, b3[ks], (short)0, acc3, false, false);
  }
  SCHED_LOADS_THEN_WMMA(20, 16);

  const float alpha = hc[head];
  const float beta  = hc[NHEAD + head];
  v8f accs[4] = {acc0, acc1, acc2, acc3};
#pragma unroll
  for (int s = 0; s < 4; ++s) {
    const int j = jt + 16 * s + l;
#pragma unroll
    for (int r = 0; r < 8; ++r) {
      int i = it + half * 8 + r;
      float sc = accs[s][r] * 0.25f + beta
               + (adjm[(size_t)i * N_NODES + j] > 0.f ? alpha : 0.f);
      S[(size_t)i * N_NODES + j] = sc;
    }
  }
}

// ---------------------------------------------------------------------------
// Top-5 + softmax + sparse attn@V for one head.
// One wave32 per row i: each lane keeps a local top5 of its strided slice,
// lane 0 merges, lanes 0..15 produce the 16 head-dim outputs.
// grid = N/8, block = 256 (8 waves = 8 rows).
// ---------------------------------------------------------------------------
__device__ __forceinline__ void top5_insert(float s, int j, float* v5, int* i5) {
#pragma unroll
  for (int t = 0; t < TOPK; ++t) {
    if (s > v5[t] || (s == v5[t] && j < i5[t])) {
#pragma unroll
      for (int u = TOPK - 1; u > t; --u) { v5[u] = v5[u - 1]; i5[u] = i5[u - 1]; }
      v5[t] = s; i5[t] = j;
      return;
    }
  }
}

__global__ void k_topk_attn(const float* __restrict__ S, const float* __restrict__ V,
                            float* __restrict__ hat, int head) {
  __shared__ float sval[8][32][TOPK];
  __shared__ int   sidx[8][32][TOPK];
  __shared__ float sp[8][TOPK];
  __shared__ int   si[8][TOPK];

  const int w    = threadIdx.x >> 5;
  const int lane = threadIdx.x & 31;
  const int i    = blockIdx.x * 8 + w;
  const float* __restrict__ row = S + (size_t)i * N_NODES;

  float v5[TOPK];
  int   i5[TOPK];
#pragma unroll
  for (int t = 0; t < TOPK; ++t) { v5[t] = -3.4e38f; i5[t] = 0x7fffffff; }
  for (int j = lane; j < N_NODES; j += 32) top5_insert(row[j], j, v5, i5);
#pragma unroll
  for (int t = 0; t < TOPK; ++t) { sval[w][lane][t] = v5[t]; sidx[w][lane][t] = i5[t]; }
  __syncthreads();

  if (lane == 0) {
    float g5[TOPK]; int gi5[TOPK];
#pragma unroll
    for (int t = 0; t < TOPK; ++t) { g5[t] = -3.4e38f; gi5[t] = 0x7fffffff; }
    for (int c = 0; c < 32; ++c)
      for (int t = 0; t < TOPK; ++t)
        top5_insert(sval[w][c][t], sidx[w][c][t], g5, gi5);
    float mx = g5[0];
    float e[TOPK], den = 0.f;
#pragma unroll
    for (int t = 0; t < TOPK; ++t) { e[t] = __expf(g5[t] - mx); den += e[t]; }
    float inv = 1.f / den;
#pragma unroll
    for (int t = 0; t < TOPK; ++t) { sp[w][t] = e[t] * inv; si[w][t] = gi5[t]; }
  }
  __syncthreads();

  if (lane < HDIM) {
    float acc = 0.f;
#pragma unroll
    for (int t = 0; t < TOPK; ++t)
      acc += sp[w][t] * V[(size_t)si[w][t] * DMODEL + head * HDIM + lane];
    hat[(size_t)i * DMODEL + head * HDIM + lane] = acc;
  }
}

// ---------------------------------------------------------------------------
// out = LayerNorm(a + b) over last dim (128). One wave per row, 4 elems/lane.
// ---------------------------------------------------------------------------
__global__ void k_add_ln(const float* __restrict__ a, const float* __restrict__ b,
                         const float* __restrict__ g, const float* __restrict__ bet,
                         float* __restrict__ out) {
  const int i = blockIdx.x;
  const int lane = threadIdx.x;
  float x[4];
  float s = 0.f;
#pragma unroll
  for (int t = 0; t < 4; ++t) {
    int d = lane + 32 * t;
    x[t] = a[(size_t)i * DMODEL + d] + b[(size_t)i * DMODEL + d];
    s += x[t];
  }
#pragma unroll
  for (int off = 16; off > 0; off >>= 1) s += __shfl_xor(s, off, 32);
  const float mean = s * (1.f / DMODEL);
  float vs = 0.f;
#pragma unroll
  for (int t = 0; t < 4; ++t) { float d = x[t] - mean; vs += d * d; }
#pragma unroll
  for (int off = 16; off > 0; off >>= 1) vs += __shfl_xor(vs, off, 32);
  const float inv = rsqrtf(vs * (1.f / DMODEL) + 1e-5f);
#pragma unroll
  for (int t = 0; t < 4; ++t) {
    int d = lane + 32 * t;
    out[(size_t)i * DMODEL + d] = g[d] * (x[t] - mean) * inv + bet[d];
  }
}

// ---------------------------------------------------------------------------
// LSTM-cell update with zero (h,c): h += sigmoid(go)*tanh(sigmoid(gi)*tanh(gg))
// gates layout [N, 4*D] as (i,f,g,o).
// ---------------------------------------------------------------------------
__global__ void k_lstm(const float* __restrict__ gates, const float* __restrict__ bhh,
                       float* __restrict__ h) {
  const int id = blockIdx.x * blockDim.x + threadIdx.x;
  if (id >= N_NODES * DMODEL) return;
  const int n = id >> 7;
  const int d = id & (DMODEL - 1);
  const float* gr = gates + (size_t)n * (4 * DMODEL);
  float gi = gr[d]               + bhh[d];
  float gg = gr[2 * DMODEL + d]  + bhh[2 * DMODEL + d];
  float go = gr[3 * DMODEL + d]  + bhh[3 * DMODEL + d];
  float c  = (1.f / (1.f + __expf(-gi))) * tanhf(gg);
  float hn = (1.f / (1.f + __expf(-go))) * tanhf(c);
  h[id] += hn;
}

// ---------------------------------------------------------------------------
// GraphNorm over node axis (per feature). One block per feature, 256 threads.
// ---------------------------------------------------------------------------
__global__ void k_graphnorm(float* __restrict__ h, const float* __restrict__ g,
                            const float* __restrict__ b) {
  const int d = blockIdx.x;
  const int t = threadIdx.x;
  float x[4], s = 0.f, s2 = 0.f;
#pragma unroll
  for (int r = 0; r < 4; ++r) {
    x[r] = h[(size_t)(t + 256 * r) * DMODEL + d];
    s += x[r]; s2 += x[r] * x[r];
  }
  __shared__ float ss[256], ss2[256];
  ss[t] = s; ss2[t] = s2;
  __syncthreads();
  for (int off = 128; off > 0; off >>= 1) {
    if (t < off) { ss[t] += ss[t + off]; ss2[t] += ss2[t + off]; }
    __syncthreads();
  }
  const float mean = ss[0] * (1.f / N_NODES);
  const float var  = ss2[0] * (1.f / N_NODES) - mean * mean;
  const float inv  = rsqrtf(var + 1e-5f);
  const float gg = g[d], bb = b[d];
#pragma unroll
  for (int r = 0; r < 4; ++r)
    h[(size_t)(t + 256 * r) * DMODEL + d] = gg * (x[r] - mean) * inv + bb;
}

__global__ void k_add(float* __restrict__ a, const float* __restrict__ b, int n) {
  int i = blockIdx.x * blockDim.x + threadIdx.x;
  if (i < n) a[i] += b[i];
}

// ---------------------------------------------------------------------------
// Host
// ---------------------------------------------------------------------------
struct LayerPtrs {
  const float *wq,*bq,*wk,*bk,*wv,*bv,*we,*be,*wo,*bo,
              *f1w,*f1b,*f2w,*f2b,*wih,*bih,*bhh,
              *n1g,*n1b,*n2g,*n2b,*gng,*gnb;
};

extern "C" void kernel_launch(void* const* d_in, const int* in_sizes, int n_in,
                              void* d_out, int out_size, void* d_ws, size_t ws_size,
                              hipStream_t stream) {
  auto F = [&](int i) { return (const float*)d_in[i]; };

  const float *nodef, *adjm, *emb_w, *emb_b, *edge_w, *edge_b, *out_w, *out_b;
  LayerPtrs lp[2];

  // Detect flatten order: dict-insertion (nodefeature first, 32768 elems)
  // vs pytree sorted-key (adjm first, 1048576 elems).
  if (in_sizes[0] == N_NODES * 32) {
    nodef = F(0); adjm = F(1);
    emb_w = F(2); emb_b = F(3); edge_w = F(4); edge_b = F(5);
    out_w = F(6); out_b = F(7);
    for (int L = 0; L < 2; ++L) {
      int o = 8 + 23 * L;
      lp[L] = LayerPtrs{ F(o+0),F(o+1),F(o+2),F(o+3),F(o+4),F(o+5),F(o+6),F(o+7),
                         F(o+8),F(o+9),F(o+10),F(o+11),F(o+12),F(o+13),F(o+14),
                         F(o+15),F(o+16),F(o+17),F(o+18),F(o+19),F(o+20),F(o+21),F(o+22) };
    }
  } else {
    adjm = F(0); nodef = F(1);
    edge_b = F(2); edge_w = F(3); emb_b = F(4); emb_w = F(5);
    for (int L = 0; L < 2; ++L) {
      int o = 6 + 23 * L;
      // sorted keys: be,bhh,bih,bk,bo,bq,bv,f1b,f1w,f2b,f2w,gnb,gng,
      //              n1b,n1g,n2b,n2g,we,wih,wk,wo,wq,wv
      lp[L].be  = F(o+0);  lp[L].bhh = F(o+1);  lp[L].bih = F(o+2);
      lp[L].bk  = F(o+3);  lp[L].bo  = F(o+4);  lp[L].bq  = F(o+5);
      lp[L].bv  = F(o+6);  lp[L].f1b = F(o+7);  lp[L].f1w = F(o+8);
      lp[L].f2b = F(o+9);  lp[L].f2w = F(o+10); lp[L].gnb = F(o+11);
      lp[L].gng = F(o+12); lp[L].n1b = F(o+13); lp[L].n1g = F(o+14);
      lp[L].n2b = F(o+15); lp[L].n2g = F(o+16); lp[L].we  = F(o+17);
      lp[L].wih = F(o+18); lp[L].wk  = F(o+19); lp[L].wo  = F(o+20);
      lp[L].wq  = F(o+21); lp[L].wv  = F(o+22);
    }
    out_b = F(52); out_w = F(53);
  }

  // Workspace carve-up (floats).
  float* w = (float*)d_ws;
  const size_t ND = (size_t)N_NODES * DMODEL;     // 131072
  float* hcur = w;            w += ND;
  float* h0   = w;            w += ND;
  float* q    = w;            w += ND;
  float* kbuf = w;            w += ND;
  float* vbuf = w;            w += ND;
  float* hat  = w;            w += ND;
  float* t128 = w;            w += ND;
  float* t512 = w;            w += (size_t)N_NODES * DFF;
  float* S    = w;            w += (size_t)N_NODES * N_NODES;
  float* hc   = w;            w += 2 * NHEAD;

  const dim3 g128(DMODEL / 64, N_NODES / 16);   // (2, 64)
  const dim3 g512(DFF / 64,    N_NODES / 16);   // (8, 64)
  const dim3 gS  (N_NODES / 64, N_NODES / 16);  // (16, 64)

  // h = nodefeature @ emb_w^T + emb_b ; h0 = h
  k_gemm_nt<<<g128, 32, 0, stream>>>(nodef, emb_w, emb_b, hcur,
                                     N_NODES, DMODEL, 32, 0);
  hipMemcpyAsync(h0, hcur, ND * sizeof(float), hipMemcpyDeviceToDevice, stream);

  for (int L = 0; L < 2; ++L) {
    const LayerPtrs& p = lp[L];
    k_headcoef<<<1, NHEAD, 0, stream>>>(p.we, p.be, edge_w, edge_b, hc);
    k_gemm_nt<<<g128, 32, 0, stream>>>(hcur, p.wq, p.bq, q,    N_NODES, DMODEL, DMODEL, 0);
    k_gemm_nt<<<g128, 32, 0, stream>>>(hcur, p.wk, p.bk, kbuf, N_NODES, DMODEL, DMODEL, 0);
    k_gemm_nt<<<g128, 32, 0, stream>>>(hcur, p.wv, p.bv, vbuf, N_NODES, DMODEL, DMODEL, 0);
    for (int hh = 0; hh < NHEAD; ++hh) {
      k_scores   <<<gS, 32, 0, stream>>>(q, kbuf, adjm, hc, S, hh);
      k_topk_attn<<<N_NODES / 8, 256, 0, stream>>>(S, vbuf, hat, hh);
    }
    k_gemm_nt<<<g128, 32, 0, stream>>>(hat, p.wo, p.bo, t128, N_NODES, DMODEL, DMODEL, 0);
    k_add_ln<<<N_NODES, 32, 0, stream>>>(hcur, t128, p.n1g, p.n1b, hcur);
    k_gemm_nt<<<g512, 32, 0, stream>>>(hcur, p.wih, p.bih, t512, N_NODES, DFF, DMODEL, 0);
    k_lstm<<<(N_NODES * DMODEL) / 256, 256, 0, stream>>>(t512, p.bhh, hcur);
    k_graphnorm<<<DMODEL, 256, 0, stream>>>(hcur, p.gng, p.gnb);
    k_gemm_nt<<<g512, 32, 0, stream>>>(hcur, p.f1w, p.f1b, t512, N_NODES, DFF, DMODEL, 1);
    k_gemm_nt<<<g128, 32, 0, stream>>>(t512, p.f2w, p.f2b, t128, N_NODES, DMODEL, DFF, 0);
    k_add_ln<<<N_NODES, 32, 0, stream>>>(hcur, t128, p.n2g, p.n2b, hcur);
    k_add<<<(N_NODES * DMODEL) / 256, 256, 0, stream>>>(hcur, h0, N_NODES * DMODEL);
  }

  k_gemm_nt<<<g128, 32, 0, stream>>>(hcur, out_w, out_b, (float*)d_out,
                                     N_NODES, DMODEL, DMODEL, 0);
}